// HybridMemory_85323820302785
// MI455X (gfx1250) — compile-verified
//
#include <hip/hip_runtime.h>
#include <math.h>

// Problem constants (match reference)
#define B_   256
#define D_   256
#define M_   65536
#define C_   1000
#define CPAD 1008          // 63 tiles of 16 columns; classF zero-padded to CPAD rows
#define INV_TEMP 20.0f     // 1/0.05
#define EPS_ 1e-6f
#define ALPHA_ 0.5f

// scatter tiling
#define GCOLS 16                       // columns per group (64B per row segment)
#define NGROUPS (D_ / GCOLS)           // 16
#define ROWS_PER_BLOCK 4096
#define BLOCKS_PER_GROUP (M_ / ROWS_PER_BLOCK)   // 16

typedef __attribute__((ext_vector_type(2))) float v2f;
typedef __attribute__((ext_vector_type(8))) float v8f;

// ---------------- init: zero accumulators (must run every call) ---------------
__global__ void init_kernel(float* __restrict__ classF, float* __restrict__ nums,
                            float* __restrict__ out) {
    int i = blockIdx.x * blockDim.x + threadIdx.x;
    if (i < CPAD * D_) classF[i] = 0.0f;   // includes zero padding rows 1000..1007
    if (i < 1024)      nums[i]   = 0.0f;
    if (i == 0)        out[0]    = 0.0f;
}

// ---------------- normalize batch features: x = feat / max(||feat||,1e-12) ---
__global__ void normalize_kernel(const float* __restrict__ feat, float* __restrict__ x) {
    const int b = blockIdx.x;
    const int t = threadIdx.x;           // 256 threads == D_
    __shared__ float red[256];
    float v = feat[b * D_ + t];
    red[t] = v * v;
    __syncthreads();
    for (int s = 128; s > 0; s >>= 1) {
        if (t < s) red[t] += red[t + s];
        __syncthreads();
    }
    float nrm = sqrtf(red[0]);
    nrm = fmaxf(nrm, 1e-12f);
    x[b * D_ + t] = v / nrm;
}

// ---------------- per-class slot counts (65536 atomics, negligible) ----------
__global__ void count_kernel(const int* __restrict__ labels, float* __restrict__ nums) {
    const int m = blockIdx.x * blockDim.x + threadIdx.x;
    if (m < M_) atomicAdd(&nums[labels[m]], 1.0f);
}

// ---------------- LDS-binned scatter-reduce of features by label -------------
// Each block owns one (column-group, row-chunk) pair: private [C_ x GCOLS] f32
// bin table in LDS, accumulated with ds_add_f32, flushed once with global
// atomics. Cuts global atomic count ~4x vs direct scatter.
__global__ void scatter_kernel(const float* __restrict__ features,
                               const int* __restrict__ labels,
                               float* __restrict__ classF) {
    __shared__ float lbin[C_ * GCOLS];               // 64000 bytes
    const int g  = blockIdx.x / BLOCKS_PER_GROUP;    // column group 0..15
    const int rb = blockIdx.x % BLOCKS_PER_GROUP;    // row chunk   0..15
    const int t  = threadIdx.x;                      // 256 threads

    for (int i = t; i < C_ * GCOLS; i += 256) lbin[i] = 0.0f;
    __syncthreads();

    const int col0   = g * GCOLS;
    const int sub    = t & (GCOLS - 1);              // column within group
    const int rlocal = t >> 4;                       // 16 rows in flight
    const int rowbase = rb * ROWS_PER_BLOCK;

    for (int r = rlocal; r < ROWS_PER_BLOCK; r += 16) {
        const int m = rowbase + r;
        const int c = labels[m];
        // streamed exactly once -> non-temporal, keep L2 for class tables
        const float v = __builtin_nontemporal_load(features + m * D_ + col0 + sub);
        atomicAdd(&lbin[c * GCOLS + sub], v);        // ds_add_f32
    }
    __syncthreads();

    // flush block-private bins to global class sums
    for (int i = t; i < C_ * GCOLS; i += 256) {
        const int c  = i / GCOLS;
        const int cc = i - c * GCOLS;
        atomicAdd(&classF[c * D_ + col0 + cc], lbin[i]);
    }
}

// ---------------- WMMA f32 GEMM: sim[b,c] = (x @ classF^T)/T / nums[c] -------
// One wave per 16x16 output tile. A = x (16xK), B = classF^T (Kx16).
// classF is zero-padded to CPAD rows, so no masking inside the WMMA loop.
__global__ void gemm_sim_kernel(const float* __restrict__ x,
                                const float* __restrict__ classF,
                                const float* __restrict__ nums,
                                float* __restrict__ sim) {
    const int wave = (blockIdx.x * blockDim.x + threadIdx.x) >> 5;  // 0..1007
    const int lane = threadIdx.x & 31;
    const int nTilesN = CPAD / 16;          // 63
    const int bt = wave / nTilesN;          // 0..15  (row tile)
    const int ct = wave % nTilesN;          // 0..62  (col tile)

    const int l15  = lane & 15;
    const int half = (lane >> 4) << 1;      // 0 for lanes 0-15, 2 for lanes 16-31
    const int row  = bt * 16 + l15;         // A-matrix row (batch index)
    const int cls  = ct * 16 + l15;         // B-matrix column (padded class index)

    v8f acc = {0.f, 0.f, 0.f, 0.f, 0.f, 0.f, 0.f, 0.f};

    const float* arow = x + row * D_;
    const float* brow = classF + cls * D_;
    for (int k0 = 0; k0 < D_; k0 += 4) {
        // 8B-aligned paired loads (k0 % 4 == 0, half in {0,2})
        v2f a = *(const v2f*)(arow + k0 + half);
        v2f b = *(const v2f*)(brow + k0 + half);
        // D = A(16x4) * B(4x16) + C   --  V_WMMA_F32_16X16X4_F32
        acc = __builtin_amdgcn_wmma_f32_16x16x4_f32(
            /*neg_a=*/false, a, /*neg_b=*/false, b,
            /*c_mod=*/(short)0, acc, /*reuse_a=*/false, /*reuse_b=*/false);
    }

    // epilogue: scale by 1/T and divide by class count (masked classes -> /1)
    const float num_c = nums[cls];          // nums padded with zeros
    const float denom = (num_c > 0.0f) ? num_c : 1.0f;
    const float scale = INV_TEMP / denom;

    // C/D layout: VGPR g -> rows (bt*16 + g) [lanes 0-15], (bt*16 + g + 8) [lanes 16-31]
    const int mbase = bt * 16 + ((lane >> 4) << 3);
    const int n = ct * 16 + l15;
#pragma unroll
    for (int g = 0; g < 8; ++g) {
        sim[(mbase + g) * CPAD + n] = acc[g] * scale;
    }
}

// ---------------- fused masked softmax + loss reduction ----------------------
__global__ void softmax_loss_kernel(const float* __restrict__ sim,
                                    const float* __restrict__ nums,
                                    const float* __restrict__ softlabels, // [M_, C_]
                                    const int* __restrict__ indexes,      // [B_]
                                    const int* __restrict__ labels,       // [M_]
                                    const int* __restrict__ bi_labels,    // [B_]
                                    const int* __restrict__ cur_epoch,    // [1]
                                    float* __restrict__ out) {
    const int b = blockIdx.x;
    const int t = threadIdx.x;           // 256 threads
    __shared__ float red[256];

    const float* srow = sim + b * CPAD;
    const int tgt = labels[indexes[b]];
    const float* soft = softlabels + (long long)bi_labels[b] * C_;

    // pass 1: masked sum of exps (reference uses raw exp, no max-shift)
    float psum = 0.0f;
    for (int c = t; c < C_; c += 256) {
        float m = (nums[c] > 0.0f) ? 1.0f : 0.0f;
        psum += expf(srow[c]) * m;
    }
    red[t] = psum;
    __syncthreads();
    for (int s = 128; s > 0; s >>= 1) {
        if (t < s) red[t] += red[t + s];
        __syncthreads();
    }
    const float denom = red[0] + EPS_;
    __syncthreads();

    // pass 2: log-probs, loss terms
    float l1 = 0.0f, l2 = 0.0f;
    for (int c = t; c < C_; c += 256) {
        float m = (nums[c] > 0.0f) ? 1.0f : 0.0f;
        float prob = (expf(srow[c]) * m) / denom;
        float logp = logf(prob + EPS_);
        l2 += -soft[c] * logp;
        if (c == tgt) l1 = -logp;
    }
    red[t] = l1;
    __syncthreads();
    for (int s = 128; s > 0; s >>= 1) {
        if (t < s) red[t] += red[t + s];
        __syncthreads();
    }
    const float l1_tot = red[0];
    __syncthreads();
    red[t] = l2;
    __syncthreads();
    for (int s = 128; s > 0; s >>= 1) {
        if (t < s) red[t] += red[t + s];
        __syncthreads();
    }
    const float l2_tot = red[0];

    if (t == 0) {
        float contrib;
        if (cur_epoch[0] == 0) {
            contrib = l1_tot / (float)B_;
        } else {
            contrib = (ALPHA_ * l1_tot + (1.0f - ALPHA_) * l2_tot) / (float)B_;
        }
        atomicAdd(out, contrib);
    }
}

// -----------------------------------------------------------------------------
extern "C" void kernel_launch(void* const* d_in, const int* in_sizes, int n_in,
                              void* d_out, int out_size, void* d_ws, size_t ws_size,
                              hipStream_t stream) {
    const float* feat        = (const float*)d_in[0];   // [B, D]
    const float* features    = (const float*)d_in[1];   // [M, D]
    const float* softlabels  = (const float*)d_in[2];   // [M, C]
    const int*   indexes     = (const int*)d_in[3];     // [B]
    const int*   labels      = (const int*)d_in[4];     // [M]
    const int*   bi_labels   = (const int*)d_in[5];     // [B]
    const int*   cur_epoch   = (const int*)d_in[6];     // [1]
    float*       out         = (float*)d_out;           // scalar

    // workspace layout (floats)
    float* ws      = (float*)d_ws;
    float* classF  = ws;                           // CPAD*D_    = 258048 (zero-padded)
    float* numsF   = classF + CPAD * D_;           // 1024 (padded)
    float* xnorm   = numsF + 1024;                 // B_*D_      = 65536
    float* sim     = xnorm + B_ * D_;              // B_*CPAD    = 258048
    (void)ws_size; (void)in_sizes; (void)n_in; (void)out_size;

    // 1) zero accumulators + output
    init_kernel<<<(CPAD * D_ + 255) / 256, 256, 0, stream>>>(classF, numsF, out);

    // 2) normalize batch features
    normalize_kernel<<<B_, 256, 0, stream>>>(feat, xnorm);

    // 3) per-class slot counts
    count_kernel<<<M_ / 256, 256, 0, stream>>>(labels, numsF);

    // 4) LDS-binned scatter-reduce of the 64 MB feature bank (dominant kernel)
    scatter_kernel<<<NGROUPS * BLOCKS_PER_GROUP, 256, 0, stream>>>(features, labels, classF);

    // 5) WMMA f32 GEMM -> class-mean similarities (16 x 63 tiles, 8 waves/block)
    gemm_sim_kernel<<<(16 * (CPAD / 16)) / 8, 256, 0, stream>>>(xnorm, classF, numsF, sim);

    // 6) fused masked softmax + losses
    softmax_loss_kernel<<<B_, 256, 0, stream>>>(sim, numsF, softlabels, indexes, labels,
                                                bi_labels, cur_epoch, out);
}